// GlobalPooling_84052509982742
// MI455X (gfx1250) — compile-verified
//
#include <hip/hip_runtime.h>
#include <stdint.h>

// ---------------------------------------------------------------------------
// GlobalPooling fused pipeline for MI455X (gfx1250, wave32, WMMA).
// Memory-bound: 3 streaming passes over the 205MB x / y arrays (~27us at
// 23.3 TB/s). Main GEMM uses V_WMMA_F32_16X16X4_F32 (full fp32 precision).
// All reductions are two-level + fixed-order -> bitwise deterministic.
//
// Round-2 fix: B fragments are fetched with explicit inline-asm ds_load_b64
// (+ one s_wait_dscnt per k-step). Round 0 spilled 128 VGPRs of hoisted
// fragments to scratch; round 1's `volatile` decayed the LDS pointer to
// generic and produced serialized flat_load ... scope:SCOPE_SYS. Inline asm
// pins the codegen: real LDS ds_loads, per-iteration, no hoist, no spill.
// ---------------------------------------------------------------------------

#define C64   64
#define MAXB  8
#define BN_EPS 1e-5f

typedef float v2f __attribute__((ext_vector_type(2)));
typedef float v8f __attribute__((ext_vector_type(8)));

// seg = searchsorted(o, row, 'right')  (branch-free scan over <=8 offsets)
__device__ __forceinline__ int seg_of(int row, const int* o, int nb) {
  int s = 0;
#pragma unroll
  for (int j = 0; j < MAXB - 1; ++j)
    s += ((j < nb - 1) && (row >= o[j])) ? 1 : 0;
  return s;
}

// ---------------------------------------------------------------------------
// Pass 1: per-segment column sums of x. Block = 256 threads = 4 row-groups
// x 64 channels (fully coalesced 256B rows). Per-thread 8 register
// accumulators (cndmask select on segment id), deterministic LDS tree over
// the 4 row-groups, block partial -> ws.
// ---------------------------------------------------------------------------
__global__ void seg_colsum_kernel(const float* __restrict__ x,
                                  const long long* __restrict__ o,
                                  int n, int nb, int chunk,
                                  float* __restrict__ part) {
  __shared__ float lds[4 * MAXB * C64];
  __shared__ int s_o[MAXB];
  const int tid = threadIdx.x;
  if (tid < MAXB) s_o[tid] = (tid < nb) ? (int)o[tid] : 0x7fffffff;
  __syncthreads();

  const int c  = tid & 63;
  const int rg = tid >> 6;  // 0..3
  float acc[MAXB];
#pragma unroll
  for (int s = 0; s < MAXB; ++s) acc[s] = 0.f;

  const int r0 = blockIdx.x * chunk;
  int r1 = r0 + chunk; if (r1 > n) r1 = n;
  for (int r = r0 + rg; r < r1; r += 4) {
    float v = x[(size_t)r * C64 + c];
    int sg = seg_of(r, s_o, nb);
#pragma unroll
    for (int s = 0; s < MAXB; ++s) acc[s] += (sg == s) ? v : 0.f;
  }
#pragma unroll
  for (int s = 0; s < MAXB; ++s) lds[(rg * MAXB + s) * C64 + c] = acc[s];
  __syncthreads();
  for (int idx = tid; idx < MAXB * C64; idx += blockDim.x) {
    float a = lds[idx] + lds[MAXB * C64 + idx] +
              lds[2 * MAXB * C64 + idx] + lds[3 * MAXB * C64 + idx];
    part[(size_t)blockIdx.x * (MAXB * C64) + idx] = a;
  }
}

__global__ void reduce_seg_kernel(const float* __restrict__ part, int nblk,
                                  int nb, float* __restrict__ segsum) {
  int idx = threadIdx.x;
  if (idx < nb * C64) {
    float a = 0.f;
    for (int b = 0; b < nblk; ++b) a += part[(size_t)b * (MAXB * C64) + idx];
    segsum[idx] = a;
  }
}

// ---------------------------------------------------------------------------
// Tiny kernel: means = segsum/count ; h = relu(means@w2^T+b2) ;
// hw = h@w1h^T + b1  (the h[seg] half of the big GEMM collapses to this
// per-segment bias). Also repack w1x into WMMA-B fragment order:
// flat = t*256 + ct*64 + lane*2 + j  ->  element W[c][k],
//   c = ct*16 + (lane&15), k = 4t + 2*(lane>>4) + j
// so the GEMM reads one 8B LDS word per lane per B-fragment.
// ---------------------------------------------------------------------------
__global__ void pooled_mlp_kernel(const float* __restrict__ segsum,
                                  const long long* __restrict__ o,
                                  const float* __restrict__ w2,
                                  const float* __restrict__ b2,
                                  const float* __restrict__ w1,
                                  const float* __restrict__ b1,
                                  int nb, float* __restrict__ hwbuf,
                                  float* __restrict__ wf0) {
  __shared__ float s_mean[MAXB * C64];
  __shared__ float s_h[MAXB * C64];
  const int tid = threadIdx.x;
  for (int idx = tid; idx < nb * C64; idx += blockDim.x) {
    int s = idx >> 6;
    float cnt = (float)(o[s] - (s ? o[s - 1] : 0ll));
    s_mean[idx] = segsum[idx] / cnt;
  }
  __syncthreads();
  for (int idx = tid; idx < nb * C64; idx += blockDim.x) {
    int s = idx >> 6, c = idx & 63;
    float a = b2[c];
    for (int k = 0; k < C64; ++k) a += s_mean[s * C64 + k] * w2[c * C64 + k];
    s_h[idx] = fmaxf(a, 0.f);
  }
  __syncthreads();
  for (int idx = tid; idx < nb * C64; idx += blockDim.x) {
    int s = idx >> 6, c = idx & 63;
    float a = b1[c];
    for (int k = 0; k < C64; ++k)
      a += s_h[s * C64 + k] * w1[c * 2 * C64 + C64 + k];
    hwbuf[idx] = a;
  }
  for (int flat = tid; flat < 4096; flat += blockDim.x) {
    int j = flat & 1, lane = (flat >> 1) & 31, ct = (flat >> 6) & 3, t = flat >> 8;
    int c = ct * 16 + (lane & 15);
    int k = t * 4 + 2 * (lane >> 4) + j;
    wf0[flat] = w1[c * 2 * C64 + k];
  }
}

// ---------------------------------------------------------------------------
// Main streaming GEMM: y = x @ w1x^T + bias[seg].  One wave owns a 16-row
// tile; 64x V_WMMA_F32_16X16X4_F32 per tile (K=64 in 16 steps, 4 c-tiles).
// STATS=true : accumulate per-channel sum / sum-of-squares (deterministic
//              two-level reduction -> per-block partial in ws).
// STATS=false: relu(y) -> d_out with BN pre-folded into weights/bias.
// ---------------------------------------------------------------------------
template <bool STATS>
__global__ __launch_bounds__(256) void fused_gemm_kernel(
    const float* __restrict__ x,
    const long long* __restrict__ o,
    const float* __restrict__ wf,
    const float* __restrict__ bias,
    int n, int nb, int ntiles,
    float* __restrict__ out) {
  __shared__ __align__(16) float s_w[4096]; // 16KB fragment-ordered weights
  __shared__ float s_b[MAXB * C64];         // per-segment bias
  __shared__ int s_o[MAXB];
  const int tid  = threadIdx.x;
  const int lane = tid & 31, wv = tid >> 5;
  const int l = lane & 15, half = lane >> 4;

  for (int idx = tid; idx < 4096; idx += blockDim.x) s_w[idx] = wf[idx];
  for (int idx = tid; idx < nb * C64; idx += blockDim.x) s_b[idx] = bias[idx];
  if (tid < MAXB) s_o[tid] = (tid < nb) ? (int)o[tid] : 0x7fffffff;
  __syncthreads();

  // Per-lane LDS byte address of this lane's B fragments; k-step t lives at
  // +t*1024, the 4 c-tiles at +0/256/512/768 (immediate ds offsets).
  const unsigned wbase = (unsigned)(uintptr_t)&s_w[0] + (unsigned)lane * 8u;

  const int nw = gridDim.x * (blockDim.x >> 5);
  const int gw = blockIdx.x * (blockDim.x >> 5) + wv;

  float p[4] = {0.f, 0.f, 0.f, 0.f};
  float q[4] = {0.f, 0.f, 0.f, 0.f};

  for (int tile = gw; tile < ntiles; tile += nw) {
    const int rowbase = tile << 4;
    int arow = rowbase + l; if (arow >= n) arow = n - 1;   // clamp tail
    const float* xrow = x + (size_t)arow * C64 + 2 * half;

    if (tile + nw < ntiles)  // prefetch next tile's 4KB (global_prefetch_b8)
      __builtin_prefetch(x + (size_t)(tile + nw) * (16 * C64) + lane * 32, 0, 0);

    v8f acc[4] = {};
#pragma unroll
    for (int t = 0; t < 16; ++t) {
      v2f a = *(const v2f*)(xrow + t * 4);   // A frag: 16x4 f32, 2 VGPRs
      v2f b0, b1, b2, b3;
      unsigned at = wbase + (unsigned)(t * 1024);
      // Four B fragments for this k-step + one wait; asm volatile pins
      // these as real per-iteration LDS loads (no hoist, no spill).
      asm volatile("ds_load_b64 %0, %4\n\t"
                   "ds_load_b64 %1, %4 offset:256\n\t"
                   "ds_load_b64 %2, %4 offset:512\n\t"
                   "ds_load_b64 %3, %4 offset:768\n\t"
                   "s_wait_dscnt 0"
                   : "=v"(b0), "=v"(b1), "=v"(b2), "=v"(b3)
                   : "v"(at));
      acc[0] = __builtin_amdgcn_wmma_f32_16x16x4_f32(false, a, false, b0,
                                                     (short)0, acc[0], false, false);
      acc[1] = __builtin_amdgcn_wmma_f32_16x16x4_f32(false, a, false, b1,
                                                     (short)0, acc[1], false, false);
      acc[2] = __builtin_amdgcn_wmma_f32_16x16x4_f32(false, a, false, b2,
                                                     (short)0, acc[2], false, false);
      acc[3] = __builtin_amdgcn_wmma_f32_16x16x4_f32(false, a, false, b3,
                                                     (short)0, acc[3], false, false);
    }

    int sg[8], rowv[8];
#pragma unroll
    for (int r = 0; r < 8; ++r) {
      rowv[r] = rowbase + r + 8 * half;      // C/D layout: VGPR r -> row r(+8)
      sg[r] = seg_of(rowv[r], s_o, nb);
    }
#pragma unroll
    for (int ct = 0; ct < 4; ++ct) {
      const int c = ct * 16 + l;             // C/D layout: lane -> column
#pragma unroll
      for (int r = 0; r < 8; ++r) {
        float v = acc[ct][r] + s_b[sg[r] * C64 + c];
        if constexpr (STATS) {
          v *= (rowv[r] < n) ? 1.f : 0.f;
          p[ct] += v;
          q[ct] += v * v;
        } else {
          if (rowv[r] < n) out[(size_t)rowv[r] * C64 + c] = fmaxf(v, 0.f);
        }
      }
    }
  }

  if constexpr (STATS) {
    __shared__ float s_rp[8 * 128];
    __shared__ float s_rq[8 * 128];
#pragma unroll
    for (int ct = 0; ct < 4; ++ct) {
      s_rp[wv * 128 + ct * 32 + lane] = p[ct];
      s_rq[wv * 128 + ct * 32 + lane] = q[ct];
    }
    __syncthreads();
    if (tid < 128) {                 // fixed-order block reduction
      int isq = tid >> 6, c = tid & 63;
      int ct = c >> 4, li = c & 15;
      const float* src = isq ? s_rq : s_rp;
      float a = 0.f;
      for (int w = 0; w < 8; ++w) {
        a += src[w * 128 + ct * 32 + li];
        a += src[w * 128 + ct * 32 + li + 16];
      }
      out[(size_t)blockIdx.x * 128 + tid] = a;   // out = stats partial buffer
    }
  }
}

__global__ void reduce_stats_kernel(const float* __restrict__ part, int nblk,
                                    float* __restrict__ stats) {
  int idx = threadIdx.x;
  if (idx < 128) {
    float a = 0.f;
    for (int b = 0; b < nblk; ++b) a += part[(size_t)b * 128 + idx];
    stats[idx] = a;
  }
}

// ---------------------------------------------------------------------------
// Finalize BN and fold into GEMM: s = gamma*rsqrt(var+eps), t = beta - mu*s;
// wfS = s_c * wf0 ; biasS = hw*s + t  => pass 3 is one GEMM + relu.
// ---------------------------------------------------------------------------
__global__ void bn_fold_kernel(const float* __restrict__ stats,
                               const float* __restrict__ gamma,
                               const float* __restrict__ beta,
                               float invN, int nb,
                               const float* __restrict__ hwbuf,
                               const float* __restrict__ wf0,
                               float* __restrict__ biasS,
                               float* __restrict__ wfS) {
  __shared__ float s_s[C64], s_t[C64];
  const int tid = threadIdx.x;
  if (tid < C64) {
    float mu  = stats[tid] * invN;
    float var = stats[C64 + tid] * invN - mu * mu;
    float sc  = gamma[tid] * rsqrtf(var + BN_EPS);
    s_s[tid] = sc;
    s_t[tid] = beta[tid] - mu * sc;
  }
  __syncthreads();
  for (int idx = tid; idx < nb * C64; idx += blockDim.x)
    biasS[idx] = hwbuf[idx] * s_s[idx & 63] + s_t[idx & 63];
  for (int flat = tid; flat < 4096; flat += blockDim.x) {
    int lane = (flat >> 1) & 31, ct = (flat >> 6) & 3;
    int c = ct * 16 + (lane & 15);
    wfS[flat] = wf0[flat] * s_s[c];
  }
}

// ---------------------------------------------------------------------------
extern "C" void kernel_launch(void* const* d_in, const int* in_sizes, int n_in,
                              void* d_out, int out_size, void* d_ws, size_t ws_size,
                              hipStream_t stream) {
  const float*     x     = (const float*)d_in[0];
  const long long* o     = (const long long*)d_in[1];
  const float*     w2    = (const float*)d_in[2];
  const float*     b2    = (const float*)d_in[3];
  const float*     w1    = (const float*)d_in[4];
  const float*     b1    = (const float*)d_in[5];
  const float*     gamma = (const float*)d_in[6];
  const float*     beta  = (const float*)d_in[7];

  const int n  = in_sizes[0] / C64;
  int nb = in_sizes[1]; if (nb > MAXB) nb = MAXB;

  // Workspace layout (floats); size grids to whatever ws_size allows.
  const size_t fixed = (size_t)3 * (MAXB * C64) + 2 * 4096 + 128;
  size_t avail = ws_size / sizeof(float);
  size_t rem = (avail > fixed) ? (avail - fixed) : 0;
  int G1 = (int)(rem / 2 / (MAXB * C64)); if (G1 > 400) G1 = 400; if (G1 < 1) G1 = 1;
  int G3 = (int)(rem / 2 / 128);          if (G3 > 512) G3 = 512; if (G3 < 1) G3 = 1;

  float* ws       = (float*)d_ws;
  float* segpart  = ws;
  float* segsum   = segpart + (size_t)G1 * (MAXB * C64);
  float* hwbuf    = segsum + MAXB * C64;
  float* wf0      = hwbuf + MAXB * C64;
  float* wfS      = wf0 + 4096;
  float* biasS    = wfS + 4096;
  float* statpart = biasS + MAXB * C64;
  float* stats    = statpart + (size_t)G3 * 128;

  const int chunk  = ((n + G1 * 4 - 1) / (G1 * 4)) * 4;
  const int ntiles = (n + 15) >> 4;

  seg_colsum_kernel<<<G1, 256, 0, stream>>>(x, o, n, nb, chunk, segpart);
  reduce_seg_kernel<<<1, 512, 0, stream>>>(segpart, G1, nb, segsum);
  pooled_mlp_kernel<<<1, 256, 0, stream>>>(segsum, o, w2, b2, w1, b1, nb, hwbuf, wf0);
  fused_gemm_kernel<true><<<G3, 256, 0, stream>>>(x, o, wf0, hwbuf, n, nb, ntiles, statpart);
  reduce_stats_kernel<<<1, 128, 0, stream>>>(statpart, G3, stats);
  bn_fold_kernel<<<1, 256, 0, stream>>>(stats, gamma, beta, 1.0f / (float)n, nb,
                                        hwbuf, wf0, biasS, wfS);
  fused_gemm_kernel<false><<<G3, 256, 0, stream>>>(x, o, wfS, biasS, n, nb, ntiles,
                                                   (float*)d_out);
}